// DunbrackWholePoseScoringModule_66073776882274
// MI455X (gfx1250) — compile-verified
//
#include <hip/hip_runtime.h>
#include <hip/hip_bf16.h>
#include <math.h>

// Problem constants (match reference)
#define NPOSE   128
#define NBLK    2048
#define APB     16
#define NBT     24
#define NBIN    36
#define N_ATOMS (NBLK * APB)          // 32768 atoms per pose
#define WG_SIZE 256
#define CHUNKS  (NBLK / WG_SIZE)      // 8 workgroups per pose

// ---- gfx1250 async global->LDS availability probes -------------------------
#if __has_builtin(__builtin_amdgcn_global_load_async_to_lds_b32)
#define HAVE_ASYNC_LDS 1
#else
#define HAVE_ASYNC_LDS 0
#endif

typedef __attribute__((address_space(1))) int GInt;
typedef __attribute__((address_space(3))) int LInt;

__device__ __forceinline__ void wait_asynccnt0() {
#if __has_builtin(__builtin_amdgcn_s_wait_asynccnt)
    __builtin_amdgcn_s_wait_asynccnt(0);
#else
    asm volatile("s_wait_asynccnt 0" ::: "memory");
#endif
}

// ---- small float3 helpers --------------------------------------------------
struct V3 { float x, y, z; };

__device__ __forceinline__ V3 ld3(const float* p) {
    V3 v; v.x = p[0]; v.y = p[1]; v.z = p[2]; return v;
}
__device__ __forceinline__ V3 vsub(V3 a, V3 b) {
    V3 r; r.x = a.x - b.x; r.y = a.y - b.y; r.z = a.z - b.z; return r;
}
__device__ __forceinline__ V3 vcross(V3 a, V3 b) {
    V3 r;
    r.x = a.y * b.z - a.z * b.y;
    r.y = a.z * b.x - a.x * b.z;
    r.z = a.x * b.y - a.y * b.x;
    return r;
}
__device__ __forceinline__ float vdot(V3 a, V3 b) {
    return a.x * b.x + a.y * b.y + a.z * b.z;
}

// signed dihedral angle, identical formulation to the reference
__device__ __forceinline__ float dihedral(V3 p0, V3 p1, V3 p2, V3 p3) {
    V3 b1 = vsub(p1, p0);
    V3 b2 = vsub(p2, p1);
    V3 b3 = vsub(p3, p2);
    V3 n1 = vcross(b1, b2);
    V3 n2 = vcross(b2, b3);
    float inv = 1.0f / (sqrtf(vdot(b2, b2)) + 1e-12f);
    V3 b2n; b2n.x = b2.x * inv; b2n.y = b2.y * inv; b2n.z = b2.z * inv;
    V3 m1 = vcross(n1, b2n);
    float x = vdot(n1, n2);
    float y = vdot(m1, n2);
    return atan2f(y, x);
}

// ---- Kernel A: per-block energy + per-workgroup partial sum ----------------
// grid = NPOSE * CHUNKS workgroups of WG_SIZE threads; thread <-> one block.
__global__ void dunbrack_score_kernel(const float* __restrict__ coords,
                                      const float* __restrict__ tables,
                                      const int*   __restrict__ block_coord_offset,
                                      const int*   __restrict__ block_types,
                                      const int*   __restrict__ bt_dihedral_atoms,
                                      float*       __restrict__ ws_partials) {
    __shared__ int   s_dih[NBT * 8];     // [NBT][2][4] local atom indices
    __shared__ float s_red[WG_SIZE];

    const int tid = threadIdx.x;

    // Stage the (tiny, hot) dihedral-atom map into LDS via gfx1250 async DMA.
    if (tid < NBT * 8) {
#if HAVE_ASYNC_LDS
        __builtin_amdgcn_global_load_async_to_lds_b32(
            (GInt*)(bt_dihedral_atoms + tid),
            (LInt*)(&s_dih[tid]),
            /*offset=*/0, /*cpol=*/0);
#else
        s_dih[tid] = bt_dihedral_atoms[tid];
#endif
    }
#if HAVE_ASYNC_LDS
    wait_asynccnt0();
#endif
    __syncthreads();

    const int wg    = blockIdx.x;           // 0 .. NPOSE*CHUNKS-1
    const int pose  = wg >> 3;              // /CHUNKS
    const int blk   = ((wg & (CHUNKS - 1)) << 8) | tid;   // 0 .. NBLK-1

    const int off = block_coord_offset[pose * NBLK + blk];
    const int bt  = block_types[pose * NBLK + blk];

    const float* pc = coords + (size_t)pose * (size_t)(N_ATOMS * 3);
    const float* tb = tables + (size_t)bt * (NBIN * NBIN);

    const float kbin = (float)(NBIN / (2.0 * M_PI));   // 36 / 2pi
    const float pif  = (float)M_PI;

    float fp = 0.f, fs = 0.f;
    int ip0 = 0, is0 = 0, ip1 = 0, is1 = 0;

    #pragma unroll
    for (int d = 0; d < 2; ++d) {
        const int* di = &s_dih[(bt * 2 + d) * 4];
        V3 q0 = ld3(pc + (size_t)(off + di[0]) * 3);
        V3 q1 = ld3(pc + (size_t)(off + di[1]) * 3);
        V3 q2 = ld3(pc + (size_t)(off + di[2]) * 3);
        V3 q3 = ld3(pc + (size_t)(off + di[3]) * 3);
        float ang = dihedral(q0, q1, q2, q3);

        float u    = (ang + pif) * kbin;
        float i0f  = floorf(u);
        float frac = u - i0f;
        int   i0   = (int)i0f;
        i0 = ((i0 % NBIN) + NBIN) % NBIN;   // safe periodic wrap
        int   i1   = (i0 + 1 == NBIN) ? 0 : i0 + 1;

        if (d == 0) { ip0 = i0; ip1 = i1; fp = frac; }
        else        { is0 = i0; is1 = i1; fs = frac; }
    }

    const float v00 = tb[ip0 * NBIN + is0];
    const float v10 = tb[ip1 * NBIN + is0];
    const float v01 = tb[ip0 * NBIN + is1];
    const float v11 = tb[ip1 * NBIN + is1];

    const float e = (1.0f - fp) * (1.0f - fs) * v00
                  +  fp          * (1.0f - fs) * v10
                  + (1.0f - fp) *  fs          * v01
                  +  fp          *  fs          * v11;

    // Deterministic fixed-order tree reduction over the workgroup.
    s_red[tid] = e;
    __syncthreads();
    #pragma unroll
    for (int s = WG_SIZE / 2; s > 0; s >>= 1) {
        if (tid < s) s_red[tid] += s_red[tid + s];
        __syncthreads();
    }
    if (tid == 0) ws_partials[wg] = s_red[0];
}

// ---- Kernel B: deterministic reduction of per-chunk partials ---------------
__global__ void dunbrack_reduce_kernel(const float* __restrict__ ws_partials,
                                       float*       __restrict__ out) {
    const int p = threadIdx.x;
    if (p < NPOSE) {
        float s = 0.0f;
        #pragma unroll
        for (int k = 0; k < CHUNKS; ++k) s += ws_partials[p * CHUNKS + k];
        out[p] = s;
    }
}

extern "C" void kernel_launch(void* const* d_in, const int* in_sizes, int n_in,
                              void* d_out, int out_size, void* d_ws, size_t ws_size,
                              hipStream_t stream) {
    (void)in_sizes; (void)n_in; (void)out_size; (void)ws_size;
    const float* coords             = (const float*)d_in[0];
    const float* tables             = (const float*)d_in[1];
    const int*   block_coord_offset = (const int*)d_in[2];
    const int*   block_types        = (const int*)d_in[3];
    const int*   bt_dihedral_atoms  = (const int*)d_in[4];
    float*       out                = (float*)d_out;
    float*       ws                 = (float*)d_ws;   // NPOSE*CHUNKS floats

    dim3 gridA(NPOSE * CHUNKS), blockA(WG_SIZE);
    dunbrack_score_kernel<<<gridA, blockA, 0, stream>>>(
        coords, tables, block_coord_offset, block_types, bt_dihedral_atoms, ws);

    dunbrack_reduce_kernel<<<1, NPOSE, 0, stream>>>(ws, out);
}